// A_weights_72533407694899
// MI455X (gfx1250) — compile-verified
//
#include <hip/hip_runtime.h>
#include <math.h>

// x: [512, 3, 32768] f32  ->  (cos(a_j)*x, sin(a_j)*x), a_j = -2*pi*f*j/30
//
// Pure streaming workload: 604 MB traffic, ~26 us floor at 23.3 TB/s HBM.
// Strategy: wave32 + float4 (b128) loads/stores, non-temporal cache hints
// (stream > 192 MB L2, zero reuse), one sincos per thread with the trig
// argument uniform per block (blockIdx.y == row index j).

typedef float v4f __attribute__((ext_vector_type(4)));

#define N_ROWS          512
#define ROW_FLOATS      (3 * 32768)          // 98304 floats per row
#define ROW_F4          (ROW_FLOATS / 4)     // 24576 float4 per row
#define VEC_PER_THREAD  4
#define THREADS         256
#define F4_PER_BLOCK    (THREADS * VEC_PER_THREAD)   // 1024
#define BLOCKS_PER_ROW  (ROW_F4 / F4_PER_BLOCK)      // 24
#define INV_FS          (1.0f / 30.0f)
#define TWO_PI          6.28318530717958647692f

__global__ __launch_bounds__(THREADS)
void zoom_phasor_kernel(const float* __restrict__ x,
                        const float* __restrict__ freq,
                        float* __restrict__ out_re,
                        float* __restrict__ out_im)
{
    const int j = blockIdx.y;                 // row index 0..511 (uniform per block)

    // a_j = 2*pi*f*(-j)*dt ; dt == 1/FS exactly in the reference
    float s, c;
    {
        const float a = -TWO_PI * freq[0] * (float)j * INV_FS;
        sincosf(a, &s, &c);                   // one accurate sincos per thread
    }

    const size_t base = (size_t)j * ROW_F4
                      + (size_t)blockIdx.x * F4_PER_BLOCK
                      + threadIdx.x;

    const v4f* __restrict__ xv = (const v4f*)x      + base;
    v4f* __restrict__ rv       = (v4f*)out_re       + base;
    v4f* __restrict__ iv       = (v4f*)out_im       + base;

    // 4 coalesced b128 loads, then 8 b128 stores; all non-temporal.
    v4f v0 = __builtin_nontemporal_load(xv + 0 * THREADS);
    v4f v1 = __builtin_nontemporal_load(xv + 1 * THREADS);
    v4f v2 = __builtin_nontemporal_load(xv + 2 * THREADS);
    v4f v3 = __builtin_nontemporal_load(xv + 3 * THREADS);

    __builtin_nontemporal_store(v0 * c, rv + 0 * THREADS);
    __builtin_nontemporal_store(v1 * c, rv + 1 * THREADS);
    __builtin_nontemporal_store(v2 * c, rv + 2 * THREADS);
    __builtin_nontemporal_store(v3 * c, rv + 3 * THREADS);

    __builtin_nontemporal_store(v0 * s, iv + 0 * THREADS);
    __builtin_nontemporal_store(v1 * s, iv + 1 * THREADS);
    __builtin_nontemporal_store(v2 * s, iv + 2 * THREADS);
    __builtin_nontemporal_store(v3 * s, iv + 3 * THREADS);
}

extern "C" void kernel_launch(void* const* d_in, const int* in_sizes, int n_in,
                              void* d_out, int out_size, void* d_ws, size_t ws_size,
                              hipStream_t stream)
{
    const float* x    = (const float*)d_in[0];   // [512, 3, 32768] f32
    const float* freq = (const float*)d_in[1];   // [1] f32
    float* out        = (float*)d_out;           // re (50331648) then im (50331648)

    const size_t half = (size_t)N_ROWS * ROW_FLOATS;   // 50331648 elements

    dim3 grid(BLOCKS_PER_ROW, N_ROWS, 1);              // (24, 512)
    zoom_phasor_kernel<<<grid, THREADS, 0, stream>>>(x, freq, out, out + half);
}